// PresetEmbedding_30305289241122
// MI455X (gfx1250) — compile-verified
//
#include <hip/hip_runtime.h>
#include <math.h>

// Problem constants (from the reference)
#define H_    256
#define T_    32
#define L_    512
#define N_    512
#define MAXC_ 64

#define WAVES   8            // waves per block (wave32)
#define NPW     64           // n-values per wave  (8 waves x 64 = all 512 n per block)
#define CH      8            // n-values per staged chunk
#define NCHUNK  (NPW / CH)   // 8 chunks, double-buffered

typedef float v4f __attribute__((ext_vector_type(4)));
typedef __attribute__((address_space(1))) int as1_int;
typedef __attribute__((address_space(3))) int as3_int;

#if defined(__gfx1250__) && __has_builtin(__builtin_amdgcn_global_load_async_to_lds_b32)
#define USE_ASYNC 1
#else
#define USE_ASYNC 0
#endif

#if USE_ASYNC
// Issue async stage of chunk+1 into the other buffer, then wait until the
// current chunk's stage (issued one iteration ago) has landed in LDS.
// Async loads complete in order per wave, so asynccnt<=1 => older one done.
#define STAGE_NEXT(chunk)                                                      \
    do {                                                                       \
        if ((chunk) + 1 < NCHUNK) {                                            \
            if (stager)                                                        \
                __builtin_amdgcn_global_load_async_to_lds_b32(                 \
                    (as1_int*)(gp0 + (size_t)((chunk) + 1) * gstep),           \
                    (as3_int*)&su[wave][((chunk) + 1) & 1][lane], 0, 0);       \
            asm volatile("s_wait_asynccnt 0x1" ::: "memory");                  \
        } else {                                                               \
            asm volatile("s_wait_asynccnt 0x0" ::: "memory");                  \
        }                                                                      \
    } while (0)
#else
#define STAGE_NEXT(chunk)                                                      \
    do {                                                                       \
        if ((chunk) + 1 < NCHUNK && stager)                                    \
            su[wave][((chunk) + 1) & 1][lane] =                                \
                *(gp0 + (size_t)((chunk) + 1) * gstep);                        \
        __syncthreads();                                                       \
    } while (0)
#endif

__global__ __launch_bounds__(256) void preset_embedding_kernel(
    const float* __restrict__ u,                // (N, L, 3)
    const float* __restrict__ w_num,            // (T*H) flat
    const float* __restrict__ cat_table,        // (n_emb, H)
    const int*   __restrict__ num_param_types,  // (L/2)
    const int*   __restrict__ easy_index,       // (MAXC*T)
    float*       __restrict__ out)              // (N, L, H)
{
    // per-wave double buffer: 8 n-values x 4 (3 comps + pad) floats
    __shared__ float su[WAVES][2][32];

    const int tid  = threadIdx.x;
    const int lane = tid & 31;
    const int wave = tid >> 5;
    const int l    = blockIdx.x;          // one block per sequence position
    const int n0   = wave * NPW;          // this wave's first n
    const int h0   = lane * 8;            // this lane's 8 output columns

    // staging map: lane -> (n_sub = lane>>2, comp = lane&3), comp 3 idle
    const int  s_     = lane >> 2;
    const int  c_     = lane & 3;
    const bool stager = (c_ < 3);
    const float* gp0  = u + ((size_t)(n0 + s_) * L_ + l) * 3 + c_;
    const size_t gstep = (size_t)CH * L_ * 3;      // floats to advance per chunk

    // ---- stage chunk 0 ----
#if USE_ASYNC
    if (stager)
        __builtin_amdgcn_global_load_async_to_lds_b32(
            (as1_int*)gp0, (as3_int*)&su[wave][0][lane], 0, 0);
#else
    if (stager) su[wave][0][lane] = *gp0;
    __syncthreads();
#endif

    // ---- positional embedding for this lane's 8 columns (once per block) ----
    float pos[8];
    {
        const float fl = (float)l;
#pragma unroll
        for (int k = 0; k < 4; ++k) {
            const int i = (h0 >> 1) + k;   // 0..127
            // 1/omega_inv = 10000^(-i/128) = exp2(-i * log2(10000)/128)
            const float ang = fl * exp2f((float)i * (-0.10380994046523007f));
            float s, c;
            sincosf(ang, &s, &c);
            pos[2 * k]     = s;            // even h -> sin
            pos[2 * k + 1] = c;            // odd  h -> cos
        }
    }

    const size_t nstride = (size_t)L_ * H_;
    float* ob = out + (size_t)l * H_ + h0 + (size_t)n0 * nstride;

    if ((l & 1) == 0) {
        // ---------- numeric row: out = 2*(u1-0.5) * w_t[type] + pos ----------
        const int t = num_param_types[l >> 1];          // block-uniform
        const float* wr = w_num + (size_t)t * H_ + h0;
        float w[8];
#pragma unroll
        for (int k = 0; k < 8; ++k) w[k] = wr[k];

        for (int chunk = 0; chunk < NCHUNK; ++chunk) {
            STAGE_NEXT(chunk);
            const float* sb = su[wave][chunk & 1];
#pragma unroll
            for (int r = 0; r < CH; ++r) {
                const float val = 2.0f * (sb[r * 4 + 1] - 0.5f);
                v4f o0, o1;
                o0.x = fmaf(val, w[0], pos[0]);
                o0.y = fmaf(val, w[1], pos[1]);
                o0.z = fmaf(val, w[2], pos[2]);
                o0.w = fmaf(val, w[3], pos[3]);
                o1.x = fmaf(val, w[4], pos[4]);
                o1.y = fmaf(val, w[5], pos[5]);
                o1.z = fmaf(val, w[6], pos[6]);
                o1.w = fmaf(val, w[7], pos[7]);
                float* op = ob + (size_t)(chunk * CH + r) * nstride;
                __builtin_nontemporal_store(o0, (v4f*)op);
                __builtin_nontemporal_store(o1, (v4f*)(op + 4));
            }
        }
    } else {
        // ---------- categorical row: out = cat_table[easy_index[round(u2*64+u0)]] + pos ----------
        for (int chunk = 0; chunk < NCHUNK; ++chunk) {
            STAGE_NEXT(chunk);
            const float* sb = su[wave][chunk & 1];
            int idxs[CH];
#pragma unroll
            for (int r = 0; r < CH; ++r) {
                const float c0 = sb[r * 4 + 0];
                const float c2 = sb[r * 4 + 2];
                const int big = (int)rintf(c2 * (float)MAXC_ + c0); // round-half-even
                idxs[r] = easy_index[big];
            }
#pragma unroll
            for (int r = 0; r < CH; ++r) {
                const float* cr = cat_table + (size_t)idxs[r] * H_ + h0;
                const v4f a = *(const v4f*)cr;
                const v4f b = *(const v4f*)(cr + 4);
                v4f o0, o1;
                o0.x = a.x + pos[0]; o0.y = a.y + pos[1];
                o0.z = a.z + pos[2]; o0.w = a.w + pos[3];
                o1.x = b.x + pos[4]; o1.y = b.y + pos[5];
                o1.z = b.z + pos[6]; o1.w = b.w + pos[7];
                float* op = ob + (size_t)(chunk * CH + r) * nstride;
                __builtin_nontemporal_store(o0, (v4f*)op);
                __builtin_nontemporal_store(o1, (v4f*)(op + 4));
            }
        }
    }
}

extern "C" void kernel_launch(void* const* d_in, const int* in_sizes, int n_in,
                              void* d_out, int out_size, void* d_ws, size_t ws_size,
                              hipStream_t stream) {
    // setup_inputs() order:
    // 0: u_in (N*L*3 f32), 1: w_num (T*H f32), 2: cat_table (n_emb*H f32),
    // 3: cat_rows (i32), 4: num_rows (i32), 5: num_param_types (i32), 6: easy_index (i32)
    const float* u               = (const float*)d_in[0];
    const float* w_num           = (const float*)d_in[1];
    const float* cat_table       = (const float*)d_in[2];
    const int*   num_param_types = (const int*)d_in[5];
    const int*   easy_index      = (const int*)d_in[6];
    float*       out             = (float*)d_out;

    dim3 grid(L_);   // one block per sequence position; waves cover all n
    preset_embedding_kernel<<<grid, 256, 0, stream>>>(
        u, w_num, cat_table, num_param_types, easy_index, out);
}